// TextEncoder_1898375545588
// MI455X (gfx1250) — compile-verified
//
#include <hip/hip_runtime.h>

typedef __attribute__((ext_vector_type(2))) float v2f;
typedef __attribute__((ext_vector_type(8))) float v8f;

// ---------------- problem constants ----------------
#define NB   64
#define LTOK 514
#define DIM  768
#define WRD  256
#define H1N  500
#define H2N  300
#define ROWS (NB * WRD)       // 16384
#define LN_EPS 1e-5f

// =====================================================================
// Kernel 1: masked segment-mean pooling
// =====================================================================
__global__ __launch_bounds__(256) void pool_kernel(
    const float* __restrict__ texts, const int* __restrict__ wstart,
    const int* __restrict__ wend, const int* __restrict__ wlen,
    float* __restrict__ pooled)
{
    const int nw = blockIdx.x;            // n*WRD + w
    const int n  = nw >> 8;               // WRD == 256
    const int w  = nw & 255;
    const int d  = threadIdx.x;           // 3 elements per thread (768/256)
    float* outp = pooled + (size_t)nw * DIM;

    if (w >= wlen[n]) {
        outp[d] = 0.0f; outp[d + 256] = 0.0f; outp[d + 512] = 0.0f;
        return;
    }
    const int s = wstart[nw], e = wend[nw];
    const float inv = 1.0f / fmaxf((float)(e - s), 1.0f);
    const float* base = texts + ((size_t)n * LTOK + 1) * DIM;  // tok = texts[:,1:-1,:]
    float a0 = 0.f, a1 = 0.f, a2 = 0.f;
    for (int t = s; t < e; ++t) {
        const float* p = base + (size_t)t * DIM;
        a0 += p[d]; a1 += p[d + 256]; a2 += p[d + 512];
    }
    outp[d]       = a0 * inv;
    outp[d + 256] = a1 * inv;
    outp[d + 512] = a2 * inv;
}

// =====================================================================
// Kernel 2: C = relu(A @ B + bias)
//   A: MxK row-major, B: KxN row-major, C: MxN row-major
// Block: 256 thr = 8 waves (4 M x 2 N), block tile 64x64.
// Each wave: two 16x16 accumulators (independent WMMA chains).
// Staging: GLOBAL_LOAD_ASYNC_TO_LDS (ASYNCcnt DMA, B transposed via
// per-lane LDS scatter addresses), double-buffered so chunk c+1's DMA
// overlaps chunk c's V_WMMA_F32_16X16X4_F32 work.
// KC divides K exactly for both GEMMs -> no K tails, EXEC all-1s at
// every WMMA; N tails: clamped loads, predicated stores.
// =====================================================================
template<int KC, int LDSTR>
__global__ __launch_bounds__(256) void gemm_bias_relu_wmma(
    const float* __restrict__ A, const float* __restrict__ B,
    const float* __restrict__ bias, float* __restrict__ C,
    int M, int N, int K)
{
    __shared__ __align__(16) float As[2][64 * LDSTR];   // [m][k]
    __shared__ __align__(16) float Bs[2][64 * LDSTR];   // [n][k] (transposed)

    const int tid  = threadIdx.x;
    const int lane = tid & 31;
    const int wave = tid >> 5;     // 0..7
    const int wm   = wave >> 1;    // 0..3 (M tile within block)
    const int wn   = wave & 1;     // 0..1 (N half within block)
    const int half = lane >> 4;    // 0 or 1 (lane group)
    const int l16  = lane & 15;

    const int n0 = blockIdx.x * 64;
    const int m0 = blockIdx.y * 64;

    const unsigned asBase = (unsigned)(uintptr_t)&As[0][0];
    const unsigned bsBase = (unsigned)(uintptr_t)&Bs[0][0];
    const unsigned bufSz  = (unsigned)(64 * LDSTR * sizeof(float));

    // ---- async staging of one K-chunk into LDS buffer `buf` ----
    auto stage = [&](int k0, int buf) {
        // A tile: 64 x KC, row-major, b128 granularity (16B aligned both sides)
        for (int f = tid; f < 64 * (KC / 4); f += 256) {
            const int row = f / (KC / 4);
            const int c4  = f % (KC / 4);
            const float* g = &A[(size_t)(m0 + row) * K + k0 + c4 * 4];
            const unsigned l = asBase + (unsigned)buf * bufSz
                             + (unsigned)((row * LDSTR + c4 * 4) * 4);
            asm volatile("global_load_async_to_lds_b128 %0, %1, off"
                         :: "v"(l), "v"(g) : "memory");
        }
        // B tile: KC x 64, transposed scatter into Bs[n][k], b32 granularity
        for (int i = tid; i < KC * 64; i += 256) {
            const int kk = i >> 6;
            const int nn = i & 63;
            int gn = n0 + nn; if (gn >= N) gn = N - 1;   // clamp; junk cols never stored
            const float* g = &B[(size_t)(k0 + kk) * N + gn];
            const unsigned l = bsBase + (unsigned)buf * bufSz
                             + (unsigned)((nn * LDSTR + kk) * 4);
            asm volatile("global_load_async_to_lds_b32 %0, %1, off"
                         :: "v"(l), "v"(g) : "memory");
        }
    };

    v8f acc0 = {}, acc1 = {};

    stage(0, 0);
    int buf = 0;
    for (int k0 = 0; k0 < K; k0 += KC) {
        asm volatile("s_wait_asynccnt 0x0" ::: "memory");  // buf's DMA complete
        __syncthreads();                                   // visible to all waves
        if (k0 + KC < K) stage(k0 + KC, buf ^ 1);          // overlap next DMA

        const float* as = &As[buf][0];
        const float* bs = &Bs[buf][0];
        #pragma unroll
        for (int k = 0; k < KC; k += 4) {
            v2f a, b0, b1;
            // A 16x4 frag: low lanes K=k,k+1 ; high lanes K=k+2,k+3
            const int ao = (wm * 16 + l16) * LDSTR + k + 2 * half;
            a.x = as[ao]; a.y = as[ao + 1];
            // B 4x16 frags for two N tiles
            const int bo = (wn * 32 + l16) * LDSTR + k + 2 * half;
            b0.x = bs[bo];             b0.y = bs[bo + 1];
            b1.x = bs[bo + 16 * LDSTR]; b1.y = bs[bo + 16 * LDSTR + 1];
            acc0 = __builtin_amdgcn_wmma_f32_16x16x4_f32(
                false, a, false, b0, (short)0, acc0, false, false);
            acc1 = __builtin_amdgcn_wmma_f32_16x16x4_f32(
                false, a, false, b1, (short)0, acc1, false, false);
        }
        __syncthreads();   // all reads of buf done before it is restaged
        buf ^= 1;
    }

    // ---- bias + relu + predicated stores (C/D layout: VGPR r -> M=r, hi lanes +8)
    const int rbase = m0 + wm * 16 + half * 8;
    const int col0  = n0 + wn * 32 + l16;
    const int col1  = col0 + 16;
    if (col0 < N) {
        const float bv = bias[col0];
        #pragma unroll
        for (int r = 0; r < 8; ++r) {
            float v = acc0[r] + bv;
            v = v > 0.0f ? v : 0.0f;
            C[(size_t)(rbase + r) * N + col0] = v;
        }
    }
    if (col1 < N) {
        const float bv = bias[col1];
        #pragma unroll
        for (int r = 0; r < 8; ++r) {
            float v = acc1[r] + bv;
            v = v > 0.0f ? v : 0.0f;
            C[(size_t)(rbase + r) * N + col1] = v;
        }
    }
}

// =====================================================================
// Kernel 3: per-row LayerNorm over H2N=300, one block (128 thr) per row
// =====================================================================
__global__ __launch_bounds__(128) void layernorm_kernel(
    const float* __restrict__ H, const float* __restrict__ gamma,
    const float* __restrict__ beta, float* __restrict__ out)
{
    const int row = blockIdx.x;
    const float* h = H + (size_t)row * H2N;

    float sum = 0.f, sumsq = 0.f;
    for (int i = threadIdx.x; i < H2N; i += 128) {
        const float v = h[i];
        sum += v; sumsq += v * v;
    }
    #pragma unroll
    for (int off = 16; off > 0; off >>= 1) {
        sum   += __shfl_down(sum,   off, 32);
        sumsq += __shfl_down(sumsq, off, 32);
    }
    __shared__ float s1[4], s2[4];
    const int lane = threadIdx.x & 31, w = threadIdx.x >> 5;
    if (lane == 0) { s1[w] = sum; s2[w] = sumsq; }
    __syncthreads();
    if (threadIdx.x == 0) {
        const float a = s1[0] + s1[1] + s1[2] + s1[3];
        const float b = s2[0] + s2[1] + s2[2] + s2[3];
        const float mu  = a / (float)H2N;
        const float var = b / (float)H2N - mu * mu;
        s1[0] = mu;
        s2[0] = rsqrtf(var + LN_EPS);
    }
    __syncthreads();
    const float mu = s1[0], inv = s2[0];
    for (int i = threadIdx.x; i < H2N; i += 128) {
        out[(size_t)row * H2N + i] = (h[i] - mu) * inv * gamma[i] + beta[i];
    }
}

// =====================================================================
extern "C" void kernel_launch(void* const* d_in, const int* in_sizes, int n_in,
                              void* d_out, int out_size, void* d_ws, size_t ws_size,
                              hipStream_t stream)
{
    const float* texts  = (const float*)d_in[0];
    const int*   wstart = (const int*)  d_in[1];
    const int*   wend   = (const int*)  d_in[2];
    const int*   wlen   = (const int*)  d_in[3];
    const float* W1     = (const float*)d_in[4];
    const float* b1     = (const float*)d_in[5];
    const float* W2     = (const float*)d_in[6];
    const float* b2     = (const float*)d_in[7];
    const float* gamma  = (const float*)d_in[8];
    const float* beta   = (const float*)d_in[9];
    float* out = (float*)d_out;

    // workspace layout (h2 reuses pooled's region once pooled is dead)
    float* pooled = (float*)d_ws;                                           // 16384x768
    float* h1 = (float*)((char*)d_ws + (size_t)ROWS * DIM * sizeof(float)); // 16384x500
    float* h2 = (float*)d_ws;                                               // 16384x300

    // 1) pooling
    pool_kernel<<<ROWS, 256, 0, stream>>>(texts, wstart, wend, wlen, pooled);

    // 2) h1 = relu(pooled @ W1 + b1)   M=16384, K=768 (= 24*32), N=500
    {
        dim3 grid((H1N + 63) / 64, ROWS / 64);
        gemm_bias_relu_wmma<32, 36><<<grid, 256, 0, stream>>>(
            pooled, W1, b1, h1, ROWS, H1N, DIM);
    }
    // 3) h2 = relu(h1 @ W2 + b2)       M=16384, K=500 (= 25*20), N=300
    {
        dim3 grid((H2N + 63) / 64, ROWS / 64);
        gemm_bias_relu_wmma<20, 24><<<grid, 256, 0, stream>>>(
            h1, W2, b2, h2, ROWS, H2N, H1N);
    }
    // 4) LayerNorm -> out
    layernorm_kernel<<<ROWS, 128, 0, stream>>>(h2, gamma, beta, out);
}